// Parallel_Linear_5334349381779
// MI455X (gfx1250) — compile-verified
//
#include <hip/hip_runtime.h>

typedef __attribute__((ext_vector_type(2))) float v2f;
typedef __attribute__((ext_vector_type(8))) float v8f;

#define C_TOTAL 64
#define C_SEL   32
#define IN_F    128
#define OUT_F   128
#define BATCH   16384

#define M_BLOCK    64            // batch rows per workgroup (4 WMMA M-tiles)
#define LDS_STRIDE (IN_F + 4)    // pad rows -> 132 floats, kills bank conflicts

__global__ __launch_bounds__(256)
void parallel_linear_wmma(const float* __restrict__ input,
                          const float* __restrict__ weight,
                          const float* __restrict__ bias,
                          const int*   __restrict__ channels,
                          float* __restrict__ out)
{
    __shared__ float smemA[M_BLOCK * LDS_STRIDE];   // 64 x 132 f32 = 33 KB

    const int tid  = threadIdx.x;
    const int lane = tid & 31;
    const int wave = tid >> 5;          // 0..7 -> which 16-wide N tile
    const int hlf  = lane >> 4;         // half-wave select (K pair / M+8)
    const int l15  = lane & 15;

    const int chSel = blockIdx.y;                 // selected-channel slot 0..31
    const int b0    = blockIdx.x * M_BLOCK;       // first batch row of block

    const int c = channels[chSel];                // gathered channel index (uniform)

    // ---- cooperative stage of A block (64 rows x 128 K, fp32) into LDS ----
    {
        const float* src = input + (size_t)b0 * (C_SEL * IN_F) + (size_t)chSel * IN_F;
        #pragma unroll
        for (int i = 0; i < 8; ++i) {
            int idx  = tid + i * 256;             // 0..2047 float4 slots
            int row  = idx >> 5;                  // 0..63
            int col4 = idx & 31;                  // float4 within row
            float4 v = *(const float4*)(src + (size_t)row * (C_SEL * IN_F) + col4 * 4);
            *(float4*)(&smemA[row * LDS_STRIDE + col4 * 4]) = v;
        }
    }
    __syncthreads();

    // ---- load this wave's 128x16 B panel into registers, kept for all M tiles.
    // B layout for V_WMMA_F32_16X16X4_F32: VGPR0 = {K0 | K2}, VGPR1 = {K1 | K3},
    // N = lane&15 (+0 / +0 across halves).  B[k][n] = W[c][n][k], so each lane
    // reads contiguous (k, k+1) pairs of its weight row -> global_load_b64.
    const float* wrow = weight + (size_t)c * (OUT_F * IN_F)
                               + (size_t)(wave * 16 + l15) * IN_F
                               + hlf * 2;
    v2f Bf[32];
    #pragma unroll
    for (int j = 0; j < 32; ++j)
        Bf[j] = *(const v2f*)(wrow + 4 * j);

    // ---- bias: C/D layout gives every acc VGPR of a lane the same N column ----
    const float bv = bias[c * OUT_F + wave * 16 + l15];

    float* outBase = out + (size_t)chSel * OUT_F + wave * 16 + l15;

    #pragma unroll
    for (int m = 0; m < M_BLOCK / 16; ++m) {
        v8f acc = {bv, bv, bv, bv, bv, bv, bv, bv};   // bias folded into init

        // A fragment: lane row = m*16 + (lane&15); lanes 0-15 read K=4j+{0,1},
        // lanes 16-31 read K=4j+{2,3}  -> ds_load_b64, conflict-free (pad=4).
        const float* arow = &smemA[(m * 16 + l15) * LDS_STRIDE + hlf * 2];

        #pragma unroll
        for (int j = 0; j < 32; ++j) {
            v2f a = *(const v2f*)(arow + 4 * j);
            acc = __builtin_amdgcn_wmma_f32_16x16x4_f32(
                /*neg_a=*/false, a, /*neg_b=*/false, Bf[j],
                /*c_mod=*/(short)0, acc, /*reuse_a=*/false, /*reuse_b=*/false);
        }

        // ---- store 16x16 f32 tile: VGPR r -> row (m*16 + hlf*8 + r), col l15 ----
        const int rbase = b0 + m * 16 + hlf * 8;
        #pragma unroll
        for (int r = 0; r < 8; ++r) {
            outBase[(size_t)(rbase + r) * (C_SEL * OUT_F)] = acc[r];
        }
    }
}

extern "C" void kernel_launch(void* const* d_in, const int* in_sizes, int n_in,
                              void* d_out, int out_size, void* d_ws, size_t ws_size,
                              hipStream_t stream) {
    (void)in_sizes; (void)n_in; (void)out_size; (void)d_ws; (void)ws_size;
    const float* input    = (const float*)d_in[0];
    const float* weight   = (const float*)d_in[1];
    const float* bias     = (const float*)d_in[2];
    const int*   channels = (const int*)d_in[3];
    float*       out      = (float*)d_out;

    dim3 grid(BATCH / M_BLOCK, C_SEL);   // (256, 32) workgroups
    parallel_linear_wmma<<<grid, 256, 0, stream>>>(input, weight, bias, channels, out);
}